// StochasticConv5_31464930410548
// MI455X (gfx1250) — compile-verified
//
#include <hip/hip_runtime.h>
#include <hip/hip_bf16.h>
#include <stdint.h>

// Problem constants (from reference)
#define BB 8
#define CC 64
#define CI 3
#define HH 224
#define WW 224
#define KK 5
#define SS 2
#define HO 110
#define WO 110

// Tiling
#define TILE_HO 8                 // output rows per block tile
#define ROWS (2*TILE_HO + 3)      // 19 input rows per tile (incl. halo)
#define NTILES ((HO + TILE_HO - 1)/TILE_HO)   // 14

#if defined(__gfx1250__) && __has_builtin(__builtin_amdgcn_tensor_load_to_lds)
#define USE_TDM 1
#else
#define USE_TDM 0
#endif

typedef uint32_t v4u __attribute__((ext_vector_type(4)));
typedef int      v8i __attribute__((ext_vector_type(8)));
typedef int      v4i __attribute__((ext_vector_type(4)));

// ---- Fallback: per-lane async global->LDS copy (16B per lane), ASYNCcnt ----
__device__ __forceinline__ void async_copy16(void* ldsp, const void* gp) {
#if defined(__gfx1250__)
    uint32_t loff = (uint32_t)(uintptr_t)ldsp;   // low 32 bits of flat shared ptr = LDS byte offset
    asm volatile("global_load_async_to_lds_b128 %0, %1, off"
                 :: "v"(loff), "v"(gp)
                 : "memory");
#else
    *(float4*)ldsp = *(const float4*)gp;
#endif
}

__device__ __forceinline__ void wait_async0() {
#if defined(__gfx1250__)
#if __has_builtin(__builtin_amdgcn_s_wait_asynccnt)
    __builtin_amdgcn_s_wait_asynccnt(0);
#else
    asm volatile("s_wait_asynccnt 0" ::: "memory");
#endif
#endif
}

__device__ __forceinline__ void wait_tensor0() {
#if defined(__gfx1250__)
#if __has_builtin(__builtin_amdgcn_s_wait_tensorcnt)
    __builtin_amdgcn_s_wait_tensorcnt(0);
#else
    asm volatile("s_wait_tensorcnt 0" ::: "memory");
#endif
#endif
}

// Block: (64, 4) = 256 threads = 8 waves (wave32).
// Grid:  (NTILES, C, B)
__global__ __launch_bounds__(256)
void grouped_conv5x5_s2_kernel(const float* __restrict__ x,
                               const float* __restrict__ w,
                               float* __restrict__ out) {
    __shared__ __align__(16) float xs[CI][ROWS][WW];   // 51,072 B
    __shared__ float wsh[CI*KK*KK];                    // 75 floats

    const int tile = blockIdx.x;                 // 0..13
    const int c    = blockIdx.y;
    const int b    = blockIdx.z;
    const int tx   = threadIdx.x;                // 0..63
    const int ty   = threadIdx.y;                // 0..3
    const int tid  = tx + 64*ty;                 // 0..255

    const int ho0 = tile * TILE_HO;
    const int r0  = ho0 * SS;                    // first input row of tile
    const int nrows = (HH - r0 < ROWS) ? (HH - r0) : ROWS;

    // Global base of this block's (b,c) input tile, ci-plane 0, row r0.
    const char* gb = (const char*)(x + (((size_t)b*CC + c) * CI) * (HH*WW) + (size_t)r0 * WW);

#if USE_TDM
    // ---- TDM: waves 0..2 each issue one 2D tensor load (nrows x 224 f32) ----
    // TDM ignores EXEC but issues once per executing wave; uniform branch gates
    // it to exactly one DMA op per ci plane.
    const int wave = tid >> 5;
    if (wave < CI) {
        const int ci = wave;
        const uint64_t ga  = (uint64_t)(uintptr_t)(gb + (size_t)ci * (HH*WW*4));
        const uint32_t lds = (uint32_t)(uintptr_t)(&xs[ci][0][0]);

        // D# group 0: count=1 | lds_addr | global_addr[56:0] | type=2
        v4u g0;
        g0.x = 1u;                                   // count=1, user descriptor
        g0.y = lds;                                  // lds_addr (bytes)
        g0.z = (uint32_t)(ga & 0xFFFFFFFFu);         // global_addr[31:0]
        g0.w = ((uint32_t)(ga >> 32) & 0x01FFFFFFu)  // global_addr[56:32]
               | 0x80000000u;                        // type = 2 ("image")

        // D# group 1: data_size=4B; tensor 224 x nrows; tile 224 x nrows (2D);
        // row stride 224 elements. tile_dim2=0 -> unused (2D copy).
        v8i g1;
        g1[0] = (int)(2u << 16);                     // workgroup_mask=0 | data_size=2 (4B)
        g1[1] = (int)(224u << 16);                   // tensor_dim0[15:0] in [31:16]
        g1[2] = (int)((uint32_t)nrows << 16);        // tensor_dim1[15:0] in [31:16]
        g1[3] = (int)(224u << 16);                   // tile_dim0 in [31:16]
        g1[4] = (int)(uint32_t)nrows;                // tile_dim1 in [15:0], tile_dim2=0
        g1[5] = 224;                                 // tensor_dim0_stride[31:0]
        g1[6] = 0;                                   // stride0[47:32]=0 | stride1[15:0]=0
        g1[7] = 0;                                   // stride1[47:16]=0

        v4i g2 = {0, 0, 0, 0};                       // dims 2/3 unused
        v4i g3 = {0, 0, 0, 0};                       // dims 3/4 unused
        v8i g4 = {0, 0, 0, 0, 0, 0, 0, 0};           // extension words (unused here)

        __builtin_amdgcn_tensor_load_to_lds(g0, g1, g2, g3, g4, 0);
        wait_tensor0();                              // this wave's DMA complete
    }
#else
    // ---- Fallback: per-lane async copies over 3 contiguous regions ----
    {
        char* lb = (char*)&xs[0][0][0];
        const int n16   = nrows * (WW*4/16);         // 16B chunks per ci region
        const int total = CI * n16;
        for (int q = tid; q < total; q += 256) {
            const int ci  = q / n16;
            const int off = (q - ci*n16) << 4;
            async_copy16(lb + (size_t)ci*(ROWS*WW*4) + off,
                         gb + (size_t)ci*(HH*WW*4) + off);
        }
        wait_async0();
    }
#endif

    // ---- Stage 75 weights for this channel into LDS (uniform per block) ----
    if (tid < CI*KK*KK) wsh[tid] = w[(size_t)c * (CI*KK*KK) + tid];

    __syncthreads();      // all waves' tile + weights visible

    // ---- Compute: each thread -> 2 (ho) x 2 (wo) outputs ----
    if (tx >= 55) return;                        // 55*2 = 110 output cols

    const int rbase = 4*ty;                      // LDS row offset for this thread's row pair
    const int colb  = 4*tx;                      // = 2*wo0 (input col base)

    float acc00 = 0.f, acc01 = 0.f, acc10 = 0.f, acc11 = 0.f;

#pragma unroll 1
    for (int ci = 0; ci < CI; ++ci) {
        float wc[25];
#pragma unroll
        for (int k = 0; k < 25; ++k) wc[k] = wsh[ci*25 + k];

#pragma unroll
        for (int r = 0; r < 7; ++r) {            // r = 2*dh + kh
            float xv[7];
#pragma unroll
            for (int j = 0; j < 7; ++j) xv[j] = xs[ci][rbase + r][colb + j];

            if (r <= 4) {                        // dh = 0, kh = r
                const int kh = r;
#pragma unroll
                for (int kw = 0; kw < 5; ++kw) {
                    acc00 = fmaf(xv[kw],     wc[kh*5 + kw], acc00);
                    acc01 = fmaf(xv[2 + kw], wc[kh*5 + kw], acc01);
                }
            }
            if (r >= 2) {                        // dh = 1, kh = r-2
                const int kh = r - 2;
#pragma unroll
                for (int kw = 0; kw < 5; ++kw) {
                    acc10 = fmaf(xv[kw],   wc[kh*5 + kw], acc10);
                    acc11 = fmaf(xv[2 + kw], wc[kh*5 + kw], acc11);
                }
            }
        }
    }

    // ---- Store (coalesced float2; index parity guarantees 8B alignment) ----
    const size_t obase = ((size_t)b*CC + c) * (size_t)(HO*WO);
    const int wo0 = 2*tx;
    const int hoA = ho0 + 2*ty;
    if (hoA < HO)
        *(float2*)(out + obase + (size_t)hoA*WO + wo0) = make_float2(acc00, acc01);
    if (hoA + 1 < HO)
        *(float2*)(out + obase + (size_t)(hoA + 1)*WO + wo0) = make_float2(acc10, acc11);
}

extern "C" void kernel_launch(void* const* d_in, const int* in_sizes, int n_in,
                              void* d_out, int out_size, void* d_ws, size_t ws_size,
                              hipStream_t stream) {
    const float* x = (const float*)d_in[0];   // [8,64,3,224,224] f32
    const float* w = (const float*)d_in[1];   // [64,3,5,5] f32
    float* out = (float*)d_out;               // [8,64,110,110] f32

    dim3 grid(NTILES, CC, BB);                // (14, 64, 8)
    dim3 block(64, 4, 1);                     // 256 threads = 8 wave32
    grouped_conv5x5_s2_kernel<<<grid, block, 0, stream>>>(x, w, out);
}